// SSIM_51178830299481
// MI455X (gfx1250) — compile-verified
//
#include <hip/hip_runtime.h>

// SSIM (window=11, sigma=1.5) on B=64, 512x512 fp32 images.
// - Separable Gaussian conv as banded matmuls on the CDNA5 fp32 matrix pipe:
//   V_WMMA_F32_16X16X4_F32, 105 WMMAs per 16x16 output tile.
// - Halo staging for interior tiles via the Tensor Data Mover
//   (tensor_load_to_lds + s_wait_tensorcnt), border tiles via predicated loads.

typedef __attribute__((ext_vector_type(2))) float v2f;
typedef __attribute__((ext_vector_type(8))) float v8f;
typedef __attribute__((ext_vector_type(4))) unsigned int u32x4;
typedef __attribute__((ext_vector_type(8))) int i32x8;
typedef __attribute__((ext_vector_type(4))) int i32x4;

#define WAVES 8
#define HDIM 512
#define WDIM 512
#define RSZ 26   // halo region rows (16 + 2*5)
#define RPAD 28  // K padded to multiple of 4 (also TDM tile width)

// gaussian(11, sigma=1.5), normalized
__constant__ float c_gw[11] = {
    0.00102838f, 0.00759877f, 0.03600078f, 0.10936069f, 0.21300420f,
    0.26601170f, 0.21300420f, 0.10936069f, 0.03600078f, 0.00759877f,
    0.00102838f};

__device__ __forceinline__ float gband(int k) {
  return (k >= 0 && k < 11) ? c_gw[k] : 0.0f;
}

__device__ __forceinline__ v8f wmma4(v2f a, v2f b, v8f c) {
  // D = A(16x4,f32) * B(4x16,f32) + C(16x16,f32)
  return __builtin_amdgcn_wmma_f32_16x16x4_f32(
      /*neg_a=*/false, a, /*neg_b=*/false, b,
      /*c_mod=*/(short)0, c, /*reuse_a=*/false, /*reuse_b=*/false);
}

// LDS byte offset of a generic pointer to __shared__ (low 32 bits of flat addr)
__device__ __forceinline__ unsigned lds_off(const void* p) {
  return (unsigned)(uintptr_t)p;
}

// TDM: DMA a 26-row x 28-col fp32 tile from a 512x512 tensor into LDS.
// gsrc must point at the tile's first element (fully in-bounds).
__device__ __forceinline__ void tdm_load_tile(const float* gsrc,
                                              unsigned lds_byte_addr) {
  const unsigned long long ga = (unsigned long long)(uintptr_t)gsrc;
  u32x4 g0;
  g0[0] = 1u;                                   // count=1, is_restore=0
  g0[1] = lds_byte_addr;                        // lds_addr [63:32]
  g0[2] = (unsigned)ga;                         // global_addr low
  g0[3] = (unsigned)((ga >> 32) & 0x1FFFFFFu)   // global_addr [56:32]
          | (2u << 30);                         // type=2 (image)
  i32x8 g1;
  g1[0] = 0x00020000;                 // data_size=2 -> 4 bytes/elem
  g1[1] = (int)(512u << 16);          // tensor_dim0[15:0] in bits[63:48]
  g1[2] = (int)(512u << 16);          // dim0 hi16=0 | tensor_dim1[15:0]
  g1[3] = (int)(28u << 16);           // tensor_dim1 hi16=0 | tile_dim0=28
  g1[4] = 26;                         // tile_dim1=26, tile_dim2=0
  g1[5] = 512;                        // tensor_dim0_stride low32
  g1[6] = 0;                          // stride hi, dim1_stride low
  g1[7] = 0;
  const i32x4 gz4 = {0, 0, 0, 0};             // groups 2/3 unused (2D tensor)
  const i32x8 gz8 = {0, 0, 0, 0, 0, 0, 0, 0}; // extra operand (clang-23 form)
  __builtin_amdgcn_tensor_load_to_lds(g0, g1, gz4, gz4, gz8, 0);
}

// signal selector: 0=p, 1=g, 2=p*p, 3=g*g, 4=p*g
template <int S>
__device__ __forceinline__ float sig(const float (*sP)[RPAD],
                                     const float (*sG)[RPAD], int r, int c) {
  if (S == 0) return sP[r][c];
  if (S == 1) return sG[r][c];
  if (S == 2) { float p = sP[r][c]; return p * p; }
  if (S == 3) { float g = sG[r][c]; return g * g; }
  return sP[r][c] * sG[r][c];
}

// Gaussian-filter one 16x16 tile of signal S. Returns WMMA C/D layout.
template <int S>
__device__ __forceinline__ v8f conv_tile(const float (*sP)[RPAD],
                                         const float (*sG)[RPAD],
                                         float (*sH)[16],
                                         const v2f* wband, int lane) {
  const int n = lane & 15;
  const int koff = (lane < 16) ? 0 : 2;
  const v8f z = {0.f, 0.f, 0.f, 0.f, 0.f, 0.f, 0.f, 0.f};

  __syncthreads();  // sH free to overwrite (prev signal's vertical reads done)

  // ---- horizontal pass: two 16-row groups covering region rows 0..25 ----
  v8f h0 = z, h1 = z;
#pragma unroll
  for (int c = 0; c < 7; ++c) {
    const int k0 = 4 * c + koff;
    v2f a0, a1;
    a0.x = sig<S>(sP, sG, n, k0);
    a0.y = sig<S>(sP, sG, n, k0 + 1);
    a1.x = sig<S>(sP, sG, 10 + n, k0);
    a1.y = sig<S>(sP, sG, 10 + n, k0 + 1);
    h0 = wmma4(a0, wband[c], h0);   // rows 0..15
    h1 = wmma4(a1, wband[c], h1);   // rows 10..25 (10..15 recomputed, identical)
  }

  // scatter H to LDS (D layout: vgpr e -> row e, +8 for lanes 16..31)
  const int mh = (lane < 16) ? 0 : 8;
#pragma unroll
  for (int e = 0; e < 8; ++e) sH[e + mh][n] = h0[e];
#pragma unroll
  for (int e = 0; e < 8; ++e) sH[10 + e + mh][n] = h1[e];

  __syncthreads();  // H visible

  // ---- vertical pass: Out = WbandT(16x28) x H(28x16) ----
  v8f v = z;
#pragma unroll
  for (int c = 0; c < 7; ++c) {
    const int k0 = 4 * c + koff;
    v2f b;
    b.x = sH[k0][n];
    b.y = sH[k0 + 1][n];
    v = wmma4(wband[c], b, v);  // vertical A == horizontal B (band symmetry)
  }
  return v;
}

__global__ __launch_bounds__(256) void ssim_tile_kernel(
    const float* __restrict__ pred, const float* __restrict__ gt,
    float* __restrict__ partial) {
  __shared__ float sP[WAVES][RSZ][RPAD];
  __shared__ float sG[WAVES][RSZ][RPAD];
  __shared__ float sH[WAVES][RPAD][16];
  __shared__ float sRed[WAVES][32];

  const int lane = threadIdx.x & 31;
  const int wave = threadIdx.x >> 5;
  const int tile = blockIdx.x * WAVES + wave;   // 64*32*32 tiles total
  const int b = tile >> 10;
  const int ty = (tile >> 5) & 31;
  const int tx = tile & 31;
  const int y0 = ty * 16, x0 = tx * 16;

  const float* pb = pred + (size_t)b * (HDIM * WDIM);
  const float* gb = gt + (size_t)b * (HDIM * WDIM);

  const bool interior = (ty >= 1) && (ty <= 30) && (tx >= 1) && (tx <= 30);
  if (interior) {
    // Whole 26x28 halo in-bounds: DMA it with the Tensor Data Mover.
    // Cols 26,27 carry real data; they only ever multiply zero band weights.
    const size_t off = (size_t)(y0 - 5) * WDIM + (x0 - 5);
    tdm_load_tile(pb + off, lds_off(&sP[wave][0][0]));
    tdm_load_tile(gb + off, lds_off(&sG[wave][0][0]));
    __builtin_amdgcn_s_wait_tensorcnt(0);
  } else {
    // Border tile: predicated staging with SAME-conv zero padding.
    for (int idx = lane; idx < RSZ * RPAD; idx += 32) {
      const int r = idx / RPAD, c = idx - r * RPAD;
      const int y = y0 - 5 + r, x = x0 - 5 + c;
      const bool ok = (c < RSZ) && ((unsigned)y < (unsigned)HDIM) &&
                      ((unsigned)x < (unsigned)WDIM);
      sP[wave][r][c] = ok ? pb[y * WDIM + x] : 0.0f;
      sG[wave][r][c] = ok ? gb[y * WDIM + x] : 0.0f;
    }
  }
  // zero-pad rows 26,27 of H once (K padding for vertical pass)
  sH[wave][26 + (lane >> 4)][lane & 15] = 0.0f;

  // banded gaussian operand; serves as horizontal B and vertical A
  v2f wband[7];
  {
    const int n = lane & 15;
    const int koff = (lane < 16) ? 0 : 2;
#pragma unroll
    for (int c = 0; c < 7; ++c) {
      const int k0 = 4 * c + koff;
      v2f w;
      w.x = gband(k0 - n);
      w.y = gband(k0 + 1 - n);
      wband[c] = w;
    }
  }

  const v8f mu1 = conv_tile<0>(sP[wave], sG[wave], sH[wave], wband, lane);
  const v8f mu2 = conv_tile<1>(sP[wave], sG[wave], sH[wave], wband, lane);
  const v8f fpp = conv_tile<2>(sP[wave], sG[wave], sH[wave], wband, lane);
  const v8f fgg = conv_tile<3>(sP[wave], sG[wave], sH[wave], wband, lane);
  const v8f fpg = conv_tile<4>(sP[wave], sG[wave], sH[wave], wband, lane);

  // pointwise SSIM on identical accumulator layouts
  const float C1 = 0.0001f, C2 = 0.0009f;
  float lsum = 0.0f;
#pragma unroll
  for (int e = 0; e < 8; ++e) {
    const float m1 = mu1[e], m2 = mu2[e];
    const float m11 = m1 * m1, m22 = m2 * m2, m12 = m1 * m2;
    const float s1 = fpp[e] - m11;
    const float s2 = fgg[e] - m22;
    const float s12 = fpg[e] - m12;
    const float num = (2.0f * m12 + C1) * (2.0f * s12 + C2);
    const float den = (m11 + m22 + C1) * (s1 + s2 + C2);
    lsum += num / den;
  }

  // deterministic per-tile sum
  sRed[wave][lane] = lsum;
  __syncthreads();
  if (lane == 0) {
    float t = 0.0f;
#pragma unroll
    for (int i = 0; i < 32; ++i) t += sRed[wave][i];
    partial[tile] = t;
  }
}

__global__ __launch_bounds__(256) void ssim_reduce_kernel(
    const float* __restrict__ partial, float* __restrict__ out, int n) {
  __shared__ float red[256];
  float s = 0.0f;
  for (int i = threadIdx.x; i < n; i += 256) s += partial[i];
  red[threadIdx.x] = s;
  __syncthreads();
  for (int off = 128; off > 0; off >>= 1) {
    if ((int)threadIdx.x < off) red[threadIdx.x] += red[threadIdx.x + off];
    __syncthreads();
  }
  if (threadIdx.x == 0)
    out[0] = 1.0f - red[0] * (1.0f / 16777216.0f);  // 64*512*512 pixels
}

extern "C" void kernel_launch(void* const* d_in, const int* in_sizes, int n_in,
                              void* d_out, int out_size, void* d_ws,
                              size_t ws_size, hipStream_t stream) {
  const float* pred = (const float*)d_in[0];
  const float* gt = (const float*)d_in[1];
  float* partial = (float*)d_ws;  // 65536 floats = 256 KB
  float* out = (float*)d_out;

  const int tiles = 64 * 32 * 32;  // 65536 16x16 tiles
  ssim_tile_kernel<<<tiles / WAVES, 256, 0, stream>>>(pred, gt, partial);
  ssim_reduce_kernel<<<1, 256, 0, stream>>>(partial, out, tiles);
}